// FastWeightsClassifier_54589034332373
// MI455X (gfx1250) — compile-verified
//
#include <hip/hip_runtime.h>
#include <hip/hip_bf16.h>

typedef __attribute__((ext_vector_type(2))) float v2f;
typedef __attribute__((ext_vector_type(8))) float v8f;

#define Bsz   64
#define Tn    64
#define En    128
#define Hn    256
#define HEADn 100
#define NCLSn 10
#define LAMc  0.9f
#define ETAc  0.5f
#define EPSc  1e-5f

// LDS row stride for A: 260 floats -> 260 % 64 == 4, so 16-lane float4 row
// reads land on disjoint 4-bank groups (conflict-free per 16-lane phase).
#define ASTRIDE 260

// ---------------------------------------------------------------------------
// Kernel 1: XU[m][n] = sum_k emb[x_ids[m]][k] * U[n][k]
//   m in [0, B*T) = 4096, n in [0, 256), k in [0, 128)
// One wave32 per 16x16 output tile, V_WMMA_F32_16X16X4_F32, K stepped by 4.
// A-operand lane layout (ISA 7.12.2): lanes 0-15 row M=lane, hold K={k,k+1};
// lanes 16-31 row M=lane-16, hold K={k+2,k+3}. B symmetric on columns.
// ---------------------------------------------------------------------------
__global__ __launch_bounds__(256) void xu_gemm_wmma(
    const int* __restrict__ x_ids, const float* __restrict__ emb,
    const float* __restrict__ U, float* __restrict__ XU) {
  const int wave  = (blockIdx.x * blockDim.x + threadIdx.x) >> 5;  // 0..4095
  const int lane  = threadIdx.x & 31;
  const int ntile = wave & 15;        // 16 N-tiles
  const int mtile = wave >> 4;        // 256 M-tiles
  const int m0 = mtile * 16;
  const int n0 = ntile * 16;
  const int lr    = lane & 15;        // row-in-tile (A) / col-in-tile (B)
  const int khalf = (lane >> 4) << 1; // 0 or 2

  const int id = x_ids[m0 + lr];
  const float* arow = emb + id * En;            // gather through embedding
  const float* brow = U + (n0 + lr) * En;       // U row-major (H,E)

  v8f c = {};
#pragma unroll 8
  for (int k = 0; k < En; k += 4) {
    v2f a = *(const v2f*)(arow + k + khalf);
    v2f b = *(const v2f*)(brow + k + khalf);
    // (neg_a, A, neg_b, B, c_mod, C, reuse_a, reuse_b)
    c = __builtin_amdgcn_wmma_f32_16x16x4_f32(
        false, a, false, b, (short)0, c, false, false);
  }

  // C/D layout: VGPR r -> M = r + 8*(lane>=16), N = lane%16
  const int rbase = m0 + ((lane >> 4) << 3);
  const int col   = n0 + lr;
#pragma unroll
  for (int r = 0; r < 8; ++r)
    XU[(rbase + r) * Hn + col] = c[r];
}

// ---------------------------------------------------------------------------
// Kernel 2: the sequential fast-weight recurrence + classifier head.
// One workgroup (256 threads = 8 wave32) per batch item; this item's full
// 256x256 fp32 fast-weight matrix A lives in LDS (CDNA5: 320KB per WGP).
// Thread i owns row i of A / element i of all H-vectors.
// ---------------------------------------------------------------------------
__global__ __launch_bounds__(256) void fastweight_rec(
    const float* __restrict__ XU, const float* __restrict__ W,
    const float* __restrict__ ln_g, const float* __restrict__ ln_b,
    const float* __restrict__ hw1, const float* __restrict__ hb1,
    const float* __restrict__ hw2, const float* __restrict__ hb2,
    float* __restrict__ out) {
  extern __shared__ float smem[];
  float* Ash    = smem;                  // 256*260 floats (~260KB)
  float* hbuf   = Ash + Hn * ASTRIDE;    // 256  : h_{t-1}
  float* hsbuf  = hbuf + Hn;             // 256  : broadcast vector
  float* hidbuf = hsbuf + Hn;            // 128  : head hidden
  float* red    = hidbuf + 128;          // 16   : reduction partials

  const int b    = blockIdx.x;
  const int i    = threadIdx.x;
  const int lane = i & 31;
  const int wid  = i >> 5;

  float* Arow = Ash + i * ASTRIDE;
  const float* Wrow = W + i * Hn;
  const float gi = ln_g[i];
  const float bi = ln_b[i];

  // Zero state (every launch: deterministic).
#pragma unroll 4
  for (int k = 0; k < Hn; k += 4)
    *(float4*)(Arow + k) = make_float4(0.f, 0.f, 0.f, 0.f);
  hbuf[i] = 0.f;
  __syncthreads();

  for (int t = 0; t < Tn; ++t) {
    // u_i = XU[b,t,i] + W[i,:] . h   (W streamed from L2, h broadcast in LDS)
    float u = XU[(b * Tn + t) * Hn + i];
#pragma unroll 8
    for (int k = 0; k < Hn; k += 4) {
      float4 w  = *(const float4*)(Wrow + k);
      float4 hh = *(const float4*)(hbuf + k);
      u += w.x * hh.x + w.y * hh.y + w.z * hh.z + w.w * hh.w;
    }
    float hs0 = fmaxf(u, 0.f);
    hsbuf[i] = hs0;
    __syncthreads();

    // Ah_i = A[i,:] . hs   (row from LDS, conflict-free float4; hs broadcast)
    float ah = 0.f;
#pragma unroll 8
    for (int k = 0; k < Hn; k += 4) {
      float4 a = *(const float4*)(Arow + k);
      float4 s = *(const float4*)(hsbuf + k);
      ah += a.x * s.x + a.y * s.y + a.z * s.z + a.w * s.w;
    }
    const float y = u + ah;

    // LayerNorm stats over H=256: wave32 shfl tree + 8-partial combine.
    float s1 = y, s2 = y * y;
#pragma unroll
    for (int off = 16; off > 0; off >>= 1) {
      s1 += __shfl_xor(s1, off, 32);
      s2 += __shfl_xor(s2, off, 32);
    }
    if (lane == 0) { red[wid] = s1; red[8 + wid] = s2; }
    __syncthreads();
    float m = 0.f, q = 0.f;
#pragma unroll
    for (int w = 0; w < 8; ++w) { m += red[w]; q += red[8 + w]; }
    m *= (1.f / Hn);
    q  = q * (1.f / Hn) - m * m;            // E[y^2] - E[y]^2
    const float hn = fmaxf(gi * (y - m) * rsqrtf(q + EPSc) + bi, 0.f);

    hsbuf[i] = hn;                          // new broadcast vector
    hbuf[i]  = hn;                          // h_t
    __syncthreads();

    // Fast-weight update: A = LAM*A + ETA * hn_i * hn_k (RMW rows in LDS)
    const float ehn = ETAc * hn;
#pragma unroll 8
    for (int k = 0; k < Hn; k += 4) {
      float4 a = *(float4*)(Arow + k);
      float4 s = *(const float4*)(hsbuf + k);
      a.x = LAMc * a.x + ehn * s.x;
      a.y = LAMc * a.y + ehn * s.y;
      a.z = LAMc * a.z + ehn * s.z;
      a.w = LAMc * a.w + ehn * s.w;
      *(float4*)(Arow + k) = a;
    }
    __syncthreads();
  }

  // Classifier head (per workgroup, h in hbuf).
  if (i < HEADn) {
    float acc = hb1[i];
    const float* w1 = hw1 + i * Hn;
#pragma unroll 8
    for (int k = 0; k < Hn; k += 4) {
      float4 w  = *(const float4*)(w1 + k);
      float4 hh = *(const float4*)(hbuf + k);
      acc += w.x * hh.x + w.y * hh.y + w.z * hh.z + w.w * hh.w;
    }
    hidbuf[i] = fmaxf(acc, 0.f);
  }
  __syncthreads();
  if (i < NCLSn) {
    float acc = hb2[i];
    const float* w2 = hw2 + i * HEADn;
    for (int k = 0; k < HEADn; ++k) acc += w2[k] * hidbuf[k];
    out[b * NCLSn + i] = acc;
  }
}

// ---------------------------------------------------------------------------
extern "C" void kernel_launch(void* const* d_in, const int* in_sizes, int n_in,
                              void* d_out, int out_size, void* d_ws, size_t ws_size,
                              hipStream_t stream) {
  const int*   x_ids = (const int*)d_in[0];
  const float* emb   = (const float*)d_in[1];
  const float* U_w   = (const float*)d_in[2];
  const float* W_w   = (const float*)d_in[3];
  const float* ln_g  = (const float*)d_in[4];
  const float* ln_b  = (const float*)d_in[5];
  const float* hw1   = (const float*)d_in[6];
  const float* hb1   = (const float*)d_in[7];
  const float* hw2   = (const float*)d_in[8];
  const float* hb2   = (const float*)d_in[9];
  float* out = (float*)d_out;
  float* XU  = (float*)d_ws;   // (B*T, H) fp32 = 4 MB scratch

  // 4096 wave-tiles = 512 blocks x 8 waves
  xu_gemm_wmma<<<512, 256, 0, stream>>>(x_ids, emb, U_w, XU);

  size_t lds_bytes = (size_t)(Hn * ASTRIDE + Hn + Hn + 128 + 16) * sizeof(float);
  fastweight_rec<<<Bsz, 256, lds_bytes, stream>>>(
      XU, W_w, ln_g, ln_b, hw1, hb1, hw2, hb2, out);
}